// PoseEstAssocEmbedLoss_14061722927137
// MI455X (gfx1250) — compile-verified
//
#include <hip/hip_runtime.h>
#include <hip/hip_bf16.h>

#define Bn 32
#define Kn 12
#define Mn 10
#define Hn 256
#define Wn 256

typedef float f32x4 __attribute__((ext_vector_type(4)));
typedef float v2f   __attribute__((ext_vector_type(2)));
typedef float v8f   __attribute__((ext_vector_type(8)));

constexpr int KHW     = Kn * Hn * Wn;   // 786432
constexpr int NTOT    = Bn * KHW;       // 25165824 (= 3 * 2^23, exact in f32)
constexpr int NVEC    = NTOT / 4;       // 6291456
constexpr int BLOCKS1 = 4096;
constexpr int THREADS1 = 256;
// BLOCKS1 * THREADS1 * 6 == NVEC exactly -> no tail, EXEC stays all-ones (WMMA requirement)

// ---- wave32 all-lane sum via V_WMMA_F32_16X16X4_F32 ----------------------
// A (16x4 f32): lane l<16 -> row l, K=0/1 in VGPR0/1; lane l>=16 -> row l-16, K=2/3.
// Put per-lane partial p in A.K0 slot, 0 in the other; B = all-ones (layout
// independent). D[m][n] = p_m + p_{m+16}. C/D layout: VGPR j holds row j
// (lanes 0-15) and row j+8 (lanes 16-31), so summing the 8 D VGPRs gives
// rowsums 0..7 in the low half-wave and 8..15 in the high half-wave; one
// xor-16 shuffle-add finishes the 32-lane total (broadcast to every lane).
__device__ __forceinline__ float wave32_sum_wmma(float p) {
    v2f a; a.x = p;   a.y = 0.0f;
    v2f b; b.x = 1.0f; b.y = 1.0f;
    v8f c = {};
    c = __builtin_amdgcn_wmma_f32_16x16x4_f32(
            /*neg_a=*/false, a, /*neg_b=*/false, b,
            /*c_mod=*/(short)0, c, /*reuse_a=*/false, /*reuse_b=*/false);
    float s = c[0] + c[1] + c[2] + c[3] + c[4] + c[5] + c[6] + c[7];
    s += __shfl_xor(s, 16, 32);
    return s;
}

// ---- kernel 1: streaming MSE partial sums (HBM-bound, ~200 MB @ 23.3 TB/s) ----
__global__ __launch_bounds__(THREADS1) void mse_partial_kernel(
        const float* __restrict__ inf,    // (B, 2K, H, W)
        const float* __restrict__ heat,   // (B, K, H, W)
        float* __restrict__ partials) {   // [BLOCKS1]
    __shared__ float wsum[THREADS1 / 32];
    const f32x4* heat4 = (const f32x4*)heat;
    const unsigned tid = threadIdx.x;
    const unsigned gid = blockIdx.x * (unsigned)THREADS1 + tid;

    float acc = 0.0f;
#pragma unroll
    for (int it = 0; it < 6; ++it) {
        unsigned v   = gid + (unsigned)it * (unsigned)(BLOCKS1 * THREADS1);
        unsigned e   = v * 4u;                      // element index, < 2^25
        unsigned bi  = e / (unsigned)KHW;           // batch
        unsigned rem = e - bi * (unsigned)KHW;      // offset within (K,H,W)
        const f32x4* ip = (const f32x4*)(inf + (size_t)bi * (size_t)(2 * KHW) + rem);
        // read-once streams: non-temporal to avoid polluting the 192MB L2
        f32x4 hv = __builtin_nontemporal_load(&heat4[v]);
        f32x4 iv = __builtin_nontemporal_load(ip);
        f32x4 d  = iv - hv;
        acc += d.x * d.x + d.y * d.y + d.z * d.z + d.w * d.w;
    }

    float ws = wave32_sum_wmma(acc);          // EXEC all-ones here by construction
    if ((tid & 31u) == 0u) wsum[tid >> 5] = ws;
    __syncthreads();
    if (tid == 0u) {
        float s = 0.0f;
#pragma unroll
        for (int w = 0; w < THREADS1 / 32; ++w) s += wsum[w];
        partials[blockIdx.x] = s;
    }
}

// ---- kernel 2: per-sample associative-embedding loss (one wave32 per sample) ----
__global__ __launch_bounds__(32) void embed_loss_kernel(
        const float* __restrict__ inf,    // (B, 2K, H, W); embed = channels K..2K-1
        const float* __restrict__ pose,   // (B, M, K, 2)
        const int*   __restrict__ cnt,    // (B,)
        float* __restrict__ out_embed) {  // [B]
    const int b    = blockIdx.x;
    const int lane = threadIdx.x;         // 0..31, wave32
    const float* embed_base = inf + (size_t)b * (size_t)(2 * KHW) + (size_t)KHW;
    const float* pose_b     = pose + (size_t)b * (Mn * Kn * 2);
    const int n_inst = cnt[b];

    float ev[Kn];
    bool  vis[Kn];
    float ref = 0.0f;
    int   viscnt = 0;
    float sq = 0.0f;
    bool  instvis = false;

    if (lane < Mn) {
#pragma unroll
        for (int k = 0; k < Kn; ++k) {
            float px = pose_b[(lane * Kn + k) * 2 + 0];
            float py = pose_b[(lane * Kn + k) * 2 + 1];
            int x = (int)rintf(px);       // jnp.round == round-half-even
            int y = (int)rintf(py);
            bool v = (x >= 0) && (x < Wn) && (y >= 0) && (y < Hn) && (lane < n_inst);
            int xc = min(max(x, 0), Wn - 1);
            int yc = min(max(y, 0), Hn - 1);
            float e = embed_base[(size_t)k * (Hn * Wn) + (size_t)yc * Wn + xc];
            e = v ? e : 0.0f;
            ev[k]  = e;
            vis[k] = v;
            viscnt += v ? 1 : 0;
            ref    += e;
        }
        ref /= (float)max(viscnt, 1);
        instvis = (viscnt > 0);
#pragma unroll
        for (int k = 0; k < Kn; ++k) {
            float d = ref - ev[k];
            sq += vis[k] ? d * d : 0.0f;
        }
    }

    unsigned long long ball = __ballot(instvis);
    int n_vis = __popcll(ball);

    // group (pull) term: sum of sq over lanes / n_vis
#pragma unroll
    for (int off = 16; off > 0; off >>= 1) sq += __shfl_xor(sq, off, 32);
    float grp = (n_vis > 0) ? sq / (float)n_vis : 0.0f;

    // separation (push) term: pairwise i<j over visible instances
    float sum_exps = 0.0f;
#pragma unroll
    for (int j = 1; j < Mn; ++j) {
        float rj = __shfl(ref, j, 32);
        bool  vj = ((ball >> j) & 1ull) != 0ull;
        if (instvis && vj && (lane < j)) {
            float d = ref - rj;
            sum_exps += expf(-0.5f * d * d);   // SIGMA = 1
        }
    }
#pragma unroll
    for (int off = 16; off > 0; off >>= 1) sum_exps += __shfl_xor(sum_exps, off, 32);
    float nf    = (float)n_vis;
    float denom = nf * (nf - 1.0f) * 0.5f + nf * 0.5f;
    float sep   = (n_vis > 1) ? sum_exps / fmaxf(denom, 1.0f) : 0.0f;

    if (lane == 0) out_embed[b] = grp * 1.0f /*GRP_W*/ + sep * 1.0f /*SEP_W*/;
}

// ---- kernel 3: deterministic final combine -------------------------------
__global__ __launch_bounds__(256) void final_combine_kernel(
        const float* __restrict__ partials,     // [BLOCKS1]
        const float* __restrict__ embed_losses, // [Bn]
        float* __restrict__ out) {
    __shared__ float red[256];
    const unsigned tid = threadIdx.x;
    float s = 0.0f;
    for (unsigned i = tid; i < (unsigned)BLOCKS1; i += 256u) s += partials[i];
    red[tid] = s;
    __syncthreads();
#pragma unroll
    for (unsigned off = 128; off > 0; off >>= 1) {
        if (tid < off) red[tid] += red[tid + off];
        __syncthreads();
    }
    if (tid == 0) {
        float pose_loss = red[0] / (float)NTOT;
        float el = 0.0f;
#pragma unroll
        for (int b = 0; b < Bn; ++b) el += embed_losses[b];
        out[0] = pose_loss * 1.0f /*POSE_W*/ + (el / (float)Bn) * 1.0f /*EMBED_W*/;
    }
}

extern "C" void kernel_launch(void* const* d_in, const int* in_sizes, int n_in,
                              void* d_out, int out_size, void* d_ws, size_t ws_size,
                              hipStream_t stream) {
    const float* inf  = (const float*)d_in[0];  // inference_tensor (B,2K,H,W)
    const float* heat = (const float*)d_in[1];  // joint_heatmaps   (B,K,H,W)
    const float* pose = (const float*)d_in[2];  // pose_instances   (B,M,K,2)
    const int*   cnt  = (const int*)d_in[3];    // instance_count   (B,)
    (void)in_sizes; (void)n_in; (void)out_size; (void)ws_size;

    float* ws       = (float*)d_ws;
    float* partials = ws;              // BLOCKS1 floats
    float* embed    = ws + BLOCKS1;    // Bn floats

    mse_partial_kernel<<<BLOCKS1, THREADS1, 0, stream>>>(inf, heat, partials);
    embed_loss_kernel<<<Bn, 32, 0, stream>>>(inf, pose, cnt, embed);
    final_combine_kernel<<<1, 256, 0, stream>>>(partials, embed, (float*)d_out);
}